// DepthModel_5308579578519
// MI455X (gfx1250) — compile-verified
//
#include <hip/hip_runtime.h>
#include <hip/hip_bf16.h>

#define FH 256
#define FW 512
#define NB 16

// ---------------- device helpers ----------------

__device__ __forceinline__ size_t pyr_index(int b, int c, int y, int x, int shift) {
    return ((size_t)(b * 3 + c) * FH + ((size_t)y << shift)) * FW + ((size_t)x << shift);
}

__device__ __forceinline__ float pyr_at(const float* __restrict__ img, int b, int c,
                                        int y, int x, int shift) {
    return img[pyr_index(b, c, y, x, shift)];
}

// _apply_disparity for a pyramid image channel: sample warped value at (y,x), level width W
__device__ __forceinline__ float warp_img_px(const float* __restrict__ img, int b, int c,
                                             int y, int x, int W, int shift, float d) {
    float Wf = (float)W;
    float xf = (float)(x + 1) + d * Wf;
    xf = fminf(fmaxf(xf, 0.0f), Wf + 1.0f);
    float x0 = floorf(xf);
    float x1 = fminf(x0 + 1.0f, Wf + 1.0f);
    int i0 = (int)x0, i1 = (int)x1;
    float p0 = (i0 >= 1 && i0 <= W) ? pyr_at(img, b, c, y, i0 - 1, shift) : 0.0f;
    float p1 = (i1 >= 1 && i1 <= W) ? pyr_at(img, b, c, y, i1 - 1, shift) : 0.0f;
    return (x1 - xf) * p0 + (xf - x0) * p1;
}

// _apply_disparity for a disparity field channel ch of disp (B,2,H,W)
__device__ __forceinline__ float warp_disp_px(const float* __restrict__ dsp, int b, int ch,
                                              int y, int x, int H, int W, float d) {
    float Wf = (float)W;
    float xf = (float)(x + 1) + d * Wf;
    xf = fminf(fmaxf(xf, 0.0f), Wf + 1.0f);
    float x0 = floorf(xf);
    float x1 = fminf(x0 + 1.0f, Wf + 1.0f);
    int i0 = (int)x0, i1 = (int)x1;
    const float* row = dsp + (((size_t)b * 2 + ch) * H + y) * (size_t)W;
    float p0 = (i0 >= 1 && i0 <= W) ? row[i0 - 1] : 0.0f;
    float p1 = (i1 >= 1 && i1 <= W) ? row[i1 - 1] : 0.0f;
    return (x1 - xf) * p0 + (xf - x0) * p1;
}

// ---------------- kernel 1: fused pointwise terms (L1 + LR + smoothness) ----------------

__global__ void pointwise_level_kernel(const float* __restrict__ left,
                                       const float* __restrict__ right,
                                       const float* __restrict__ disp,
                                       int H, int W, int shift,
                                       float w_l1, float w_lr, float w_sm,
                                       float* __restrict__ partials) {
    __shared__ float red[256];
    int tid = threadIdx.x;
    int p = blockIdx.x * 256 + tid;
    int b = blockIdx.y;
    float acc = 0.0f;
    if (p < H * W) {
        int y = p / W;
        int x = p - y * W;
        size_t dbase = (((size_t)b * 2) * H + y) * (size_t)W + x;
        float dl = disp[dbase];
        float dr = disp[dbase + (size_t)H * W];

        // L1 photometric: left_est = warp(right, -dl) vs left; right_est = warp(left, +dr) vs right
        float l1 = 0.0f;
        for (int c = 0; c < 3; ++c) {
            float le = warp_img_px(right, b, c, y, x, W, shift, -dl);
            l1 += fabsf(le - pyr_at(left, b, c, y, x, shift));
            float re = warp_img_px(left, b, c, y, x, W, shift, dr);
            l1 += fabsf(re - pyr_at(right, b, c, y, x, shift));
        }

        // LR consistency: r2l = warp(dr, -dl) vs dl; l2r = warp(dl, dr) vs dr
        float r2l = warp_disp_px(disp, b, 1, y, x, H, W, -dl);
        float l2r = warp_disp_px(disp, b, 0, y, x, H, W, dr);
        float lr = fabsf(r2l - dl) + fabsf(l2r - dr);

        // edge-aware smoothness (x-gradient only, valid for x < W-1)
        float sm = 0.0f;
        if (x < W - 1) {
            float gl = 0.0f, gr = 0.0f;
            for (int c = 0; c < 3; ++c) {
                gl += fabsf(pyr_at(left, b, c, y, x, shift) - pyr_at(left, b, c, y, x + 1, shift));
                gr += fabsf(pyr_at(right, b, c, y, x, shift) - pyr_at(right, b, c, y, x + 1, shift));
            }
            float wxl = __expf(-gl * (1.0f / 3.0f));
            float wxr = __expf(-gr * (1.0f / 3.0f));
            float dl1 = disp[dbase + 1];
            float dr1 = disp[dbase + (size_t)H * W + 1];
            sm = fabsf((dl - dl1) * wxl) + fabsf((dr - dr1) * wxr);
        }
        acc = w_l1 * l1 + w_lr * lr + w_sm * sm;
    }
    // deterministic block reduction
    red[tid] = acc;
    __syncthreads();
    for (int off = 128; off > 0; off >>= 1) {
        if (tid < off) red[tid] += red[tid + off];
        __syncthreads();
    }
    if (tid == 0) partials[(size_t)blockIdx.y * gridDim.x + blockIdx.x] = red[0];
}

// ---------------- kernel 2: LDS-tiled SSIM with async global->LDS staging ----------------

#define TW 32
#define TH 8
#define HW_ (TW + 2)
#define HH_ (TH + 2)

__global__ void ssim_level_kernel(const float* __restrict__ left,
                                  const float* __restrict__ right,
                                  const float* __restrict__ disp,
                                  int H, int W, int shift,
                                  float w_ss, float* __restrict__ partials) {
    __shared__ float sX[HH_ * HW_];  // warped image tile (+1 halo)
    __shared__ float sY[HH_ * HW_];  // original image tile (+1 halo), async-loaded
    int tid = threadIdx.x;
    int b = blockIdx.z;
    int tx0 = blockIdx.x * TW;
    int ty0 = blockIdx.y * TH;
    int PH = H - 2, PW = W - 2;
    int lx = tid & 31;
    int ly = tid >> 5;
    int ox = tx0 + lx;
    int oy = ty0 + ly;
    float acc = 0.0f;

    for (int side = 0; side < 2; ++side) {
        const float* src = (side == 0) ? right : left;  // warp source
        const float* org = (side == 0) ? left : right;  // comparison target
        int dch = (side == 0) ? 0 : 1;
        float sgn = (side == 0) ? -1.0f : 1.0f;
        for (int c = 0; c < 3; ++c) {
            // stage halo tile: sX via VALU gathers, sY via async global->LDS DMA
            for (int idx = tid; idx < HH_ * HW_; idx += 256) {
                int hy = idx / HW_;
                int hx = idx - hy * HW_;
                int iy = min(ty0 + hy, H - 1);
                int ix = min(tx0 + hx, W - 1);
                float d = disp[(((size_t)b * 2 + dch) * H + iy) * (size_t)W + ix] * sgn;
                sX[idx] = warp_img_px(src, b, c, iy, ix, W, shift, d);
                const float* gp = &org[pyr_index(b, c, iy, ix, shift)];
                unsigned lo = (unsigned)(size_t)&sY[idx];  // low 32 bits of flat ptr = LDS offset
                asm volatile("global_load_async_to_lds_b32 %0, %1, off"
                             :: "v"(lo), "v"(gp) : "memory");
            }
            asm volatile("s_wait_asynccnt 0" ::: "memory");
            __syncthreads();

            if (ox < PW && oy < PH) {
                float Sx = 0.f, Sy = 0.f, Sxx = 0.f, Syy = 0.f, Sxy = 0.f;
#pragma unroll
                for (int dy = 0; dy < 3; ++dy)
#pragma unroll
                    for (int dx = 0; dx < 3; ++dx) {
                        float xv = sX[(ly + dy) * HW_ + (lx + dx)];
                        float yv = sY[(ly + dy) * HW_ + (lx + dx)];
                        Sx += xv; Sy += yv;
                        Sxx += xv * xv; Syy += yv * yv; Sxy += xv * yv;
                    }
                const float inv9 = 1.0f / 9.0f, C1 = 1e-4f, C2 = 9e-4f;
                float mx = Sx * inv9, my = Sy * inv9;
                float vx = Sxx * inv9 - mx * mx;
                float vy = Syy * inv9 - my * my;
                float vxy = Sxy * inv9 - mx * my;
                float n = (2.0f * mx * my + C1) * (2.0f * vxy + C2);
                float dden = (mx * mx + my * my + C1) * (vx + vy + C2);
                float v = (1.0f - n / dden) * 0.5f;
                v = fminf(fmaxf(v, 0.0f), 1.0f);
                acc += w_ss * v;
            }
            __syncthreads();
        }
    }
    // block reduction (reuse sX)
    sX[tid] = acc;
    __syncthreads();
    for (int off = 128; off > 0; off >>= 1) {
        if (tid < off) sX[tid] += sX[tid + off];
        __syncthreads();
    }
    if (tid == 0)
        partials[((size_t)blockIdx.z * gridDim.y + blockIdx.y) * gridDim.x + blockIdx.x] = sX[0];
}

// ---------------- kernel 3: deterministic final reduction, WMMA finish ----------------

typedef __attribute__((ext_vector_type(2))) float v2f;
typedef __attribute__((ext_vector_type(8))) float v8f;

__global__ void final_reduce_kernel(const float* __restrict__ partials, int n,
                                    float* __restrict__ out) {
    int lane = threadIdx.x;  // exactly one wave32
    float s = 0.0f;
    for (int k = lane; k < n; k += 32) s += partials[k];  // fixed lane assignment: deterministic
    // Cross-lane sum via V_WMMA_F32_16X16X4_F32:
    // A[m][0]=s_m (lanes 0-15 vgpr0), A[m][2]=s_{m+16} (lanes 16-31 vgpr0), others 0.
    // B = all ones  =>  D[m][n] = s_m + s_{m+16}.
    v2f a; a.x = s; a.y = 0.0f;
    v2f bm; bm.x = 1.0f; bm.y = 1.0f;
    v8f cz = {};
    v8f d = __builtin_amdgcn_wmma_f32_16x16x4_f32(false, a, false, bm, (short)0, cz, false, false);
    float t = d[0] + d[1] + d[2] + d[3] + d[4] + d[5] + d[6] + d[7];
    float total = __shfl(t, 0, 32) + __shfl(t, 16, 32);
    if (lane == 0) out[0] = total;
}

// ---------------- host launch ----------------

extern "C" void kernel_launch(void* const* d_in, const int* in_sizes, int n_in,
                              void* d_out, int out_size, void* d_ws, size_t ws_size,
                              hipStream_t stream) {
    const float* left = (const float*)d_in[0];
    const float* right = (const float*)d_in[1];
    const float* disp[4] = {(const float*)d_in[2], (const float*)d_in[3],
                            (const float*)d_in[4], (const float*)d_in[5]};
    float* partials = (float*)d_ws;
    const float ALPHA = 0.85f;
    int off = 0;

    // fused pointwise terms per level
    for (int i = 0; i < 4; ++i) {
        int H = FH >> i, W = FW >> i;
        int bpi = (H * W + 255) / 256;
        float w_l1 = (1.0f - ALPHA) / ((float)NB * 3.0f * (float)H * (float)W);
        float w_lr = 1.0f / ((float)NB * (float)H * (float)W);
        float w_sm = (0.1f / (float)(1 << i)) / ((float)NB * (float)H * (float)(W - 1));
        pointwise_level_kernel<<<dim3(bpi, NB), 256, 0, stream>>>(
            left, right, disp[i], H, W, i, w_l1, w_lr, w_sm, partials + off);
        off += bpi * NB;
    }
    // SSIM per level
    for (int i = 0; i < 4; ++i) {
        int H = FH >> i, W = FW >> i;
        int PH = H - 2, PW = W - 2;
        int tX = (PW + TW - 1) / TW, tY = (PH + TH - 1) / TH;
        float w_ss = ALPHA / ((float)NB * 3.0f * (float)PH * (float)PW);
        ssim_level_kernel<<<dim3(tX, tY, NB), 256, 0, stream>>>(
            left, right, disp[i], H, W, i, w_ss, partials + off);
        off += tX * tY * NB;
    }
    // deterministic final sum -> scalar loss
    final_reduce_kernel<<<1, 32, 0, stream>>>(partials, off, (float*)d_out);
}